// LSTM_11879879542513
// MI455X (gfx1250) — compile-verified
//
#include <hip/hip_runtime.h>

typedef __attribute__((ext_vector_type(2))) float v2f;
typedef __attribute__((ext_vector_type(8))) float v8f;
typedef int v4i __attribute__((vector_size(16)));   // matches builtin param type

#define B_DIM 64
#define H_DIM 2048
#define BH (B_DIM * H_DIM)
#define NT 64          // N tile per workgroup
#define KC 32          // K chunk depth staged in LDS
#define NCHUNKS (H_DIM / KC)   // 64
#define AS_STRIDE 36   // padded LDS row stride for A (floats)
#define BS_STRIDE 80   // padded LDS row stride for B (floats)
#define TPB 128        // 4 waves (wave32)

#if defined(__HIP_DEVICE_COMPILE__) && \
    __has_builtin(__builtin_amdgcn_global_load_async_to_lds_b128) && \
    __has_builtin(__builtin_amdgcn_s_wait_asynccnt)
#define USE_ASYNC 1
#else
#define USE_ASYNC 0
#endif

#if USE_ASYNC
#define GLOBAL_AS __attribute__((address_space(1)))
#define LDS_AS    __attribute__((address_space(3)))
#endif

// 16-byte global -> LDS transfer (async DMA on gfx1250, sync fallback otherwise)
__device__ __forceinline__ void copy16_to_lds(const float* g, float* l) {
#if USE_ASYNC
    __builtin_amdgcn_global_load_async_to_lds_b128(
        (GLOBAL_AS v4i*)g, (LDS_AS v4i*)l, 0, 0);
#else
    *(float4*)l = *(const float4*)g;
#endif
}
__device__ __forceinline__ void async_wait_keep8() {
#if USE_ASYNC
    __builtin_amdgcn_s_wait_asynccnt(8);   // current chunk's 8 ops retired
#endif
}
__device__ __forceinline__ void async_wait_zero() {
#if USE_ASYNC
    __builtin_amdgcn_s_wait_asynccnt(0);
#endif
}

// Single-pass GEMM: dst[64 x NT tile] = A @ W (+ bias)
// blockIdx.y: 0..3 gate h-pass (+bias) -> ws[g], 4: y -> d_out,
//             5..8 gate x-pass -> ws[4+g]
__global__ __launch_bounds__(TPB) void lstm_gemm_kernel(
    const float* __restrict__ x, const float* __restrict__ h_prev,
    const float* __restrict__ Wfh, const float* __restrict__ Wfx, const float* __restrict__ bf,
    const float* __restrict__ Wih, const float* __restrict__ Wix, const float* __restrict__ bi,
    const float* __restrict__ Woh, const float* __restrict__ Wox, const float* __restrict__ bo,
    const float* __restrict__ Wch, const float* __restrict__ Wcx, const float* __restrict__ bc,
    const float* __restrict__ Wy,  const float* __restrict__ by,
    float* __restrict__ gates_ws, float* __restrict__ out)
{
    __shared__ float As[2][B_DIM * AS_STRIDE];   // 2 x 64 x 36 floats
    __shared__ float Bs[2][KC * BS_STRIDE];      // 2 x 32 x 80 floats

    const int p     = blockIdx.y;
    const int ntile = blockIdx.x;
    const int tid   = threadIdx.x;
    const int lane  = tid & 31;
    const int wave  = tid >> 5;
    const int l16   = lane & 15;
    const int kh    = lane >> 4;     // 0: lanes 0-15, 1: lanes 16-31

    const float *A = x, *W = Wox, *bias = nullptr;
    float* dst = gates_ws + 7 * BH;
    int ldo = H_DIM;
    switch (p) {
        case 0: A = h_prev; W = Wfh; bias = bf; dst = gates_ws + 0 * BH; break;
        case 1: A = h_prev; W = Wih; bias = bi; dst = gates_ws + 1 * BH; break;
        case 2: A = h_prev; W = Wch; bias = bc; dst = gates_ws + 2 * BH; break;
        case 3: A = h_prev; W = Woh; bias = bo; dst = gates_ws + 3 * BH; break;
        case 4: A = h_prev; W = Wy;  bias = by; dst = out; ldo = 3 * H_DIM; break;
        case 5: A = x; W = Wfx; dst = gates_ws + 4 * BH; break;
        case 6: A = x; W = Wix; dst = gates_ws + 5 * BH; break;
        case 7: A = x; W = Wcx; dst = gates_ws + 6 * BH; break;
        default: break;              // p == 8: x-pass of o gate (defaults)
    }

    const int ncol0 = ntile * NT;

    // issue one K-chunk's staging (8 x b128 per thread => 8 async ops per wave)
    auto issue = [&](int t, int buf) {
        const int kc = t * KC;
        #pragma unroll
        for (int i = 0; i < 4; ++i) {          // A: 64 x 32 floats
            const int idx = tid + i * TPB;
            const int row = idx >> 3;
            const int kq  = idx & 7;
            copy16_to_lds(A + (size_t)row * H_DIM + kc + kq * 4,
                          &As[buf][row * AS_STRIDE + kq * 4]);
        }
        #pragma unroll
        for (int i = 0; i < 4; ++i) {          // B: 32 x 64 floats
            const int idx = tid + i * TPB;
            const int row = idx >> 4;
            const int nq  = idx & 15;
            copy16_to_lds(W + (size_t)(kc + row) * H_DIM + ncol0 + nq * 4,
                          &Bs[buf][row * BS_STRIDE + nq * 4]);
        }
    };

    v8f acc[4];
    #pragma unroll
    for (int m = 0; m < 4; ++m)
        #pragma unroll
        for (int j = 0; j < 8; ++j) acc[m][j] = 0.0f;

    const int bcol = wave * 16 + l16;

    issue(0, 0);                                // prologue: chunk 0 -> buf 0
    for (int t = 0; t < NCHUNKS; ++t) {
        const int cur = t & 1;
        if (t + 1 < NCHUNKS) {
            issue(t + 1, cur ^ 1);              // prefetch next chunk (overlaps compute)
            async_wait_keep8();                 // chunk t fully in LDS
        } else {
            async_wait_zero();
        }
        __syncthreads();

        // ---- compute: 8 K-steps of V_WMMA_F32_16X16X4_F32 over 4 M-frags
        #pragma unroll
        for (int ks = 0; ks < 8; ++ks) {
            const int kb = ks * 4 + kh * 2;     // fragment k base for this half-wave
            v2f b;
            b.x = Bs[cur][kb * BS_STRIDE + bcol];
            b.y = Bs[cur][(kb + 1) * BS_STRIDE + bcol];
            #pragma unroll
            for (int m = 0; m < 4; ++m) {
                const v2f a = *(const v2f*)(&As[cur][(m * 16 + l16) * AS_STRIDE + kb]);
                acc[m] = __builtin_amdgcn_wmma_f32_16x16x4_f32(
                    false, a, false, b, (short)0, acc[m], false, false);
            }
        }
        __syncthreads();                        // done reading buf cur before it is refilled
    }

    // ---- epilogue: add bias, store C fragments (row = mtile*16 + vgpr + 8*half)
    const int gcol = ncol0 + wave * 16 + l16;
    const float bv = bias ? bias[gcol] : 0.0f;
    #pragma unroll
    for (int m = 0; m < 4; ++m) {
        #pragma unroll
        for (int v = 0; v < 8; ++v) {
            const int row = m * 16 + kh * 8 + v;
            dst[(size_t)row * ldo + gcol] = acc[m][v] + bv;
        }
    }
}

// Elementwise LSTM epilogue: sum h/x partial gates -> (h, c) into d_out next to y.
__global__ void lstm_pointwise_kernel(const float* __restrict__ gates,
                                      const float* __restrict__ c_prev,
                                      float* __restrict__ out)
{
    const int idx = blockIdx.x * blockDim.x + threadIdx.x;
    if (idx >= BH) return;
    const int row = idx >> 11;           // / 2048
    const int col = idx & (H_DIM - 1);

    const float f_  = gates[0 * BH + idx] + gates[4 * BH + idx];
    const float i_  = gates[1 * BH + idx] + gates[5 * BH + idx];
    const float ct_ = gates[2 * BH + idx] + gates[6 * BH + idx];
    const float o_  = gates[3 * BH + idx] + gates[7 * BH + idx];

    const float f  = 1.0f / (1.0f + expf(-f_));
    const float i  = 1.0f / (1.0f + expf(-i_));
    const float ct = tanhf(ct_);
    const float o  = 1.0f / (1.0f + expf(-o_));

    const float c = f * c_prev[idx] + i * ct;
    const float h = o * tanhf(c);

    out[(size_t)row * (3 * H_DIM) + H_DIM     + col] = h;
    out[(size_t)row * (3 * H_DIM) + 2 * H_DIM + col] = c;
}

extern "C" void kernel_launch(void* const* d_in, const int* in_sizes, int n_in,
                              void* d_out, int out_size, void* d_ws, size_t ws_size,
                              hipStream_t stream)
{
    (void)in_sizes; (void)n_in; (void)out_size; (void)ws_size;
    const float* x      = (const float*)d_in[0];
    const float* h_prev = (const float*)d_in[1];
    const float* c_prev = (const float*)d_in[2];
    const float* Wfh    = (const float*)d_in[3];
    const float* Wfx    = (const float*)d_in[4];
    const float* bf     = (const float*)d_in[5];
    const float* Wih    = (const float*)d_in[6];
    const float* Wix    = (const float*)d_in[7];
    const float* bi     = (const float*)d_in[8];
    const float* Woh    = (const float*)d_in[9];
    const float* Wox    = (const float*)d_in[10];
    const float* bo     = (const float*)d_in[11];
    const float* Wch    = (const float*)d_in[12];
    const float* Wcx    = (const float*)d_in[13];
    const float* bc     = (const float*)d_in[14];
    const float* Wy     = (const float*)d_in[15];
    const float* by     = (const float*)d_in[16];

    float* gates = (float*)d_ws;               // 8 * 64 * 2048 floats = 4 MB
    float* out   = (float*)d_out;              // [64, 6144] = y | h | c

    dim3 grid(H_DIM / NT, 9);                  // 32 N-tiles x 9 single-pass GEMMs = 288 WGs
    lstm_gemm_kernel<<<grid, TPB, 0, stream>>>(
        x, h_prev, Wfh, Wfx, bf, Wih, Wix, bi, Woh, Wox, bo,
        Wch, Wcx, bc, Wy, by, gates, out);

    lstm_pointwise_kernel<<<(BH + 255) / 256, 256, 0, stream>>>(gates, c_prev, out);
}